// SASRec_78254304133887
// MI455X (gfx1250) — compile-verified
//
#include <hip/hip_runtime.h>
#include <hip/hip_bf16.h>

// GGNN GRU cell, fully fused, f32 WMMA (V_WMMA_F32_16X16X4_F32), wave32.
// One workgroup (256 thr = 8 waves) per batch element b. All GEMMs on WMMA.
// w_ih staged by the Tensor Data Mover (TENSOR_LOAD_TO_LDS, TENSORcnt, with
// D# padding producing the banked LDS stride); w_hh staged with
// GLOBAL_LOAD_ASYNC_TO_LDS_B128 (ASYNCcnt). Both overlap phases 0-1.

typedef __attribute__((ext_vector_type(2))) float v2f;
typedef __attribute__((ext_vector_type(8))) float v8f;
typedef __attribute__((ext_vector_type(4))) unsigned int v4u;
typedef __attribute__((ext_vector_type(4))) int v4i;
typedef __attribute__((ext_vector_type(8))) int v8i;

#define HT_LD 204   // transposed h_in/h_out stride (204 % 64 = 12 -> conflict free)
#define IN_LD 130   // inputs buffer stride         (130 % 64 = 2)
#define WIH_LD 132  // w_ih LDS stride = 128 + 4 (TDM pad: interval 128 DW, amount 4 DW)
#define WHH_LD 68   // w_hh LDS stride (68 % 64 = 4; 16B aligned rows)
// LDS floats: region1 = 2*64*HT_LD (reused as 200*IN_LD <= it), + w_ih + w_hh
#define WIH_OFF (2 * 64 * HT_LD)
#define WHH_OFF (WIH_OFF + 192 * WIH_LD)
#define LDS_FLOATS (WHH_OFF + 192 * WHH_LD)

__device__ __forceinline__ v8f wmma4(v2f a, v2f b, v8f c) {
  // D(16x16,f32) = A(16x4,f32) * B(4x16,f32) + C
  return __builtin_amdgcn_wmma_f32_16x16x4_f32(false, a, false, b, (short)0, c,
                                               false, false);
}

// Async global->LDS copy, 16B per lane. lds_byte_off is the byte offset of the
// destination within the workgroup LDS allocation (dynamic LDS starts at 0).
__device__ __forceinline__ void async_g2l_b128(unsigned int lds_byte_off,
                                               const float* g) {
  asm volatile("global_load_async_to_lds_b128 %0, %1, off"
               :: "v"(lds_byte_off),
                  "v"((unsigned long long)(uintptr_t)g)
               : "memory");
}
__device__ __forceinline__ void wait_async0() {
  asm volatile("s_wait_asynccnt 0" ::: "memory");
}

// TDM: DMA a 2D tile (dim1 rows x dim0 4-byte elements, row-major, stride =
// dim0) from global into LDS at lds_byte_off, inserting `pad` DWORDs of LDS
// padding every 2^(ivl+1) DWORDs (D# group1 pad fields, ISA 8.4).
__device__ __forceinline__ void tdm_load_2d(unsigned int lds_byte_off,
                                            const void* gsrc,
                                            unsigned int dim0, unsigned int dim1,
                                            unsigned int ivl, unsigned int amt) {
  unsigned long long ga = (unsigned long long)(uintptr_t)gsrc;
  // D# group 0: count=1 | lds_addr | global_addr[56:0] | type=2 ("image")
  v4u g0 = { 1u,
             lds_byte_off,
             (unsigned int)ga,
             (unsigned int)(ga >> 32) | (2u << 30) };
  // D# group 1: data_size=2 (4B), pad_enable, pad_interval, pad_amount,
  // tensor_dim0/1 = tile_dim0/1 = full tile, tensor_dim0_stride = dim0.
  v8i g1 = { (int)((2u << 16) | (1u << 20) | (ivl << 22) | (amt << 25)),
             (int)(dim0 << 16),                    // tensor_dim0[15:0] @ 63:48
             (int)((dim0 >> 16) | (dim1 << 16)),   // tensor_dim0 hi, dim1 lo
             (int)((dim1 >> 16) | (dim0 << 16)),   // tensor_dim1 hi, tile_dim0
             (int)(dim1 & 0xFFFFu),                // tile_dim1 (tile_dim2=0)
             (int)dim0,                            // tensor_dim0_stride[31:0]
             0, 0 };
  v4i gz = { 0, 0, 0, 0 };
#if __has_include(<hip/amd_detail/amd_gfx1250_TDM.h>)
  v8i gz8 = { 0, 0, 0, 0, 0, 0, 0, 0 };
  __builtin_amdgcn_tensor_load_to_lds(g0, g1, gz, gz, gz8, 0);
#else
  __builtin_amdgcn_tensor_load_to_lds(g0, g1, gz, gz, 0);
#endif
}

// ---- Phase 0: h_in/h_out = hidden @ W^T + bias, stored TRANSPOSED in LDS ----
__device__ __forceinline__ void htile(int m0, int lr, int lh,
                                      const float* __restrict__ hB,
                                      const float* __restrict__ Win,
                                      const float* __restrict__ Wout,
                                      const float* __restrict__ bin,
                                      const float* __restrict__ bout,
                                      float* shT, float* shTo) {
  int arow = m0 + lr; if (arow > 199) arow = 199;       // clamp tail tile
  v2f ah[16];
  const float* hR = hB + arow * 64 + 2 * lh;
  #pragma unroll
  for (int kk = 0; kk < 16; ++kk) ah[kk] = *(const v2f*)(hR + 4 * kk);

  v8f zero = {};
  #pragma unroll
  for (int nt = 0; nt < 4; ++nt) {
    v8f aI = zero, aO = zero;
    // B[k][n] = W[n_global][k]  (einsum bnh,gh->bng), row-major W (64x64)
    const float* wiB = Win + (nt * 16 + lr) * 64 + 2 * lh;
    const float* woB = Wout + (nt * 16 + lr) * 64 + 2 * lh;
    #pragma unroll
    for (int kk = 0; kk < 16; ++kk) {
      aI = wmma4(ah[kk], *(const v2f*)(wiB + 4 * kk), aI);
      aO = wmma4(ah[kk], *(const v2f*)(woB + 4 * kk), aO);
    }
    float bi = bin[nt * 16 + lr], bo = bout[nt * 16 + lr];
    #pragma unroll
    for (int r = 0; r < 8; ++r) {
      int row = m0 + lh * 8 + r;
      if (row < 200) {  // transposed store: shT[col][row]
        shT [(nt * 16 + lr) * HT_LD + row] = aI[r] + bi;
        shTo[(nt * 16 + lr) * HT_LD + row] = aO[r] + bo;
      }
    }
  }
}

// ---- Phase 1: input_in/out tile = A_in/out(16x200) @ h_in/out(200x64) ----
__device__ __forceinline__ void graph_tile(int m0, int lr, int lh,
                                           const float* __restrict__ Ain,
                                           const float* __restrict__ Aout,
                                           const float* shT, const float* shTo,
                                           const float* __restrict__ biah,
                                           const float* __restrict__ boah,
                                           v8f (&oI)[4], v8f (&oO)[4]) {
  int arow = m0 + lr; if (arow > 199) arow = 199;
  const float* aiR = Ain + arow * 200 + 2 * lh;
  const float* aoR = Aout + arow * 200 + 2 * lh;
  v8f zero = {};
  #pragma unroll
  for (int n = 0; n < 4; ++n) { oI[n] = zero; oO[n] = zero; }
  #pragma unroll 2
  for (int k0 = 0; k0 < 200; k0 += 4) {
    v2f aI = *(const v2f*)(aiR + k0);
    v2f aO = *(const v2f*)(aoR + k0);
    #pragma unroll
    for (int n = 0; n < 4; ++n) {
      v2f bI = *(const v2f*)(shT  + (n * 16 + lr) * HT_LD + k0 + 2 * lh);
      v2f bO = *(const v2f*)(shTo + (n * 16 + lr) * HT_LD + k0 + 2 * lh);
      oI[n] = wmma4(aI, bI, oI[n]);
      oO[n] = wmma4(aO, bO, oO[n]);
    }
  }
  #pragma unroll
  for (int n = 0; n < 4; ++n) {
    float bi = biah[n * 16 + lr], bo = boah[n * 16 + lr];
    #pragma unroll
    for (int r = 0; r < 8; ++r) { oI[n][r] += bi; oO[n][r] += bo; }
  }
}

// ---- C-layout accumulators -> inputs(200x128) buffer in LDS ----
__device__ __forceinline__ void store_inputs(int m0, int lr, int lh,
                                             const v8f (&aI)[4],
                                             const v8f (&aO)[4],
                                             float* shInp) {
  #pragma unroll
  for (int n = 0; n < 4; ++n)
    #pragma unroll
    for (int r = 0; r < 8; ++r) {
      int row = m0 + lh * 8 + r;
      if (row < 200) {
        shInp[row * IN_LD + n * 16 + lr]      = aI[n][r];
        shInp[row * IN_LD + 64 + n * 16 + lr] = aO[n][r];
      }
    }
}

// ---- Phase 2: gi/gh + GRU gating ----
__device__ __forceinline__ void gru_tile(int b, int m0, int lr, int lh,
                                         const float* shInp,
                                         const float* shWih,
                                         const float* shWhh,
                                         const float* __restrict__ hidden,
                                         const float* __restrict__ bih,
                                         const float* __restrict__ bhh,
                                         float* __restrict__ outp) {
  int arow = m0 + lr; if (arow > 199) arow = 199;
  v2f ah[16];
  const float* hR = hidden + ((size_t)b * 200 + arow) * 64 + 2 * lh;
  #pragma unroll
  for (int kk = 0; kk < 16; ++kk) ah[kk] = *(const v2f*)(hR + 4 * kk);
  const float* inR = shInp + arow * IN_LD + 2 * lh;

  v8f zero = {};
  v8f rg[4], ig[4];
  // reset / input gates: gi_g + gh_g accumulate into one set
  #pragma unroll
  for (int g = 0; g < 2; ++g) {
    const int g0 = g * 64;
    v8f acc[4] = {zero, zero, zero, zero};
    #pragma unroll 2
    for (int k0 = 0; k0 < 128; k0 += 4) {
      v2f a = *(const v2f*)(inR + k0);
      #pragma unroll
      for (int n = 0; n < 4; ++n) {
        v2f w = *(const v2f*)(shWih + (g0 + n * 16 + lr) * WIH_LD + k0 + 2 * lh);
        acc[n] = wmma4(a, w, acc[n]);
      }
    }
    #pragma unroll
    for (int kk = 0; kk < 16; ++kk)
      #pragma unroll
      for (int n = 0; n < 4; ++n) {
        v2f w = *(const v2f*)(shWhh + (g0 + n * 16 + lr) * WHH_LD + 4 * kk + 2 * lh);
        acc[n] = wmma4(ah[kk], w, acc[n]);
      }
    #pragma unroll
    for (int n = 0; n < 4; ++n) {
      float bb = bih[g0 + n * 16 + lr] + bhh[g0 + n * 16 + lr];
      #pragma unroll
      for (int r = 0; r < 8; ++r) {
        float s = 1.0f / (1.0f + __expf(-(acc[n][r] + bb)));
        if (g == 0) rg[n][r] = s; else ig[n][r] = s;
      }
    }
  }
  // new gate: needs i_n and h_n separately (newgate = tanh(i_n + r*h_n))
  const int g0 = 128;
  v8f accA[4] = {zero, zero, zero, zero};
  v8f accH[4] = {zero, zero, zero, zero};
  #pragma unroll 2
  for (int k0 = 0; k0 < 128; k0 += 4) {
    v2f a = *(const v2f*)(inR + k0);
    #pragma unroll
    for (int n = 0; n < 4; ++n) {
      v2f w = *(const v2f*)(shWih + (g0 + n * 16 + lr) * WIH_LD + k0 + 2 * lh);
      accA[n] = wmma4(a, w, accA[n]);
    }
  }
  #pragma unroll
  for (int kk = 0; kk < 16; ++kk)
    #pragma unroll
    for (int n = 0; n < 4; ++n) {
      v2f w = *(const v2f*)(shWhh + (g0 + n * 16 + lr) * WHH_LD + 4 * kk + 2 * lh);
      accH[n] = wmma4(ah[kk], w, accH[n]);
    }
  #pragma unroll
  for (int n = 0; n < 4; ++n) {
    float bA = bih[g0 + n * 16 + lr];
    float bH = bhh[g0 + n * 16 + lr];
    #pragma unroll
    for (int r = 0; r < 8; ++r) {
      int row = m0 + lh * 8 + r;
      if (row < 200) {
        size_t off = ((size_t)b * 200 + row) * 64 + n * 16 + lr;
        float hv = hidden[off];
        float ng = tanhf((accA[n][r] + bA) + rg[n][r] * (accH[n][r] + bH));
        outp[off] = hv - ig[n][r] * (hv - ng);
      }
    }
  }
}

__global__ void __launch_bounds__(256)
ggnn_fused(const float* __restrict__ A, const float* __restrict__ hidden,
           const float* __restrict__ wih, const float* __restrict__ whh,
           const float* __restrict__ bih, const float* __restrict__ bhh,
           const float* __restrict__ biah, const float* __restrict__ boah,
           const float* __restrict__ Win, const float* __restrict__ bin,
           const float* __restrict__ Wout, const float* __restrict__ bout,
           float* __restrict__ outp) {
  extern __shared__ float smem[];
  float* shT   = smem;                        // h_inT  [64][HT_LD]
  float* shTo  = smem + 64 * HT_LD;           // h_outT [64][HT_LD]
  float* shWih = smem + WIH_OFF;              // [192][WIH_LD]
  float* shWhh = smem + WHH_OFF;              // [192][WHH_LD]
  float* shInp = smem;                        // reused: inputs [200][IN_LD]

  const int tid = threadIdx.x;
  const int lane = tid & 31, wave = tid >> 5;
  const int lr = lane & 15, lh = lane >> 4;
  const int b = blockIdx.x;

  // ---- issue weight staging (overlaps with phases 0-1) ----
  if (wave == 0) {
    // w_ih 192x128 f32 via TDM: pad 4 DWORDs (amt code 3) every 128 DWORDs
    // (ivl code 6) -> LDS row stride 132 floats == WIH_LD.
    tdm_load_2d((unsigned int)(WIH_OFF * 4), wih, 128u, 192u, 6u, 3u);
  }
  // w_hh: 192 rows x 64 floats = 16 x b128 chunks per row, async to LDS
  for (int c = tid; c < 192 * 16; c += 256) {
    int row = c >> 4, col4 = (c & 15) * 4;
    async_g2l_b128((unsigned int)((WHH_OFF + row * WHH_LD + col4) * 4),
                   whh + row * 64 + col4);
  }

  const float* hB  = hidden + (size_t)b * 200 * 64;
  const float* Ain = A + (size_t)b * 400 * 200;
  const float* Aout = Ain + 200 * 200;

  const int t0 = wave;          // tiles 0..7
  const int t1 = wave + 8;      // tiles 8..12 (waves 0..4)

  // warm L2/L0 for the dominant A stream of the second tile
  if (t1 < 13) {
    int arow1 = t1 * 16 + lr; if (arow1 > 199) arow1 = 199;
    __builtin_prefetch(Ain + arow1 * 200, 0, 3);
    __builtin_prefetch(Aout + arow1 * 200, 0, 3);
  }

  // Phase 0: build transposed h_in/h_out in LDS
  htile(t0 * 16, lr, lh, hB, Win, Wout, bin, bout, shT, shTo);
  if (t1 < 13) htile(t1 * 16, lr, lh, hB, Win, Wout, bin, bout, shT, shTo);
  __syncthreads();

  // Phase 1: A_in/out @ h_in/out, results in registers
  v8f gI0[4], gO0[4], gI1[4], gO1[4];
  graph_tile(t0 * 16, lr, lh, Ain, Aout, shT, shTo, biah, boah, gI0, gO0);
  if (t1 < 13) graph_tile(t1 * 16, lr, lh, Ain, Aout, shT, shTo, biah, boah, gI1, gO1);

  wait_async0();                                   // w_hh async done (per wave)
  if (wave == 0) __builtin_amdgcn_s_wait_tensorcnt(0);  // w_ih TDM done
  __syncthreads();   // publish weights; shT/shTo region now reusable

  // stash concatenated inputs into LDS for A-fragment access
  store_inputs(t0 * 16, lr, lh, gI0, gO0, shInp);
  if (t1 < 13) store_inputs(t1 * 16, lr, lh, gI1, gO1, shInp);
  __syncthreads();

  // Phase 2: gates + GRU update
  gru_tile(b, t0 * 16, lr, lh, shInp, shWih, shWhh, hidden, bih, bhh, outp);
  if (t1 < 13) gru_tile(b, t1 * 16, lr, lh, shInp, shWih, shWhh, hidden, bih, bhh, outp);
}

extern "C" void kernel_launch(void* const* d_in, const int* in_sizes, int n_in,
                              void* d_out, int out_size, void* d_ws, size_t ws_size,
                              hipStream_t stream) {
  const float* A        = (const float*)d_in[0];
  const float* hidden   = (const float*)d_in[1];
  const float* w_ih     = (const float*)d_in[2];
  const float* w_hh     = (const float*)d_in[3];
  const float* b_ih     = (const float*)d_in[4];
  const float* b_hh     = (const float*)d_in[5];
  const float* b_iah    = (const float*)d_in[6];
  const float* b_oah    = (const float*)d_in[7];
  const float* W_in     = (const float*)d_in[8];
  const float* bias_in  = (const float*)d_in[9];
  const float* W_out    = (const float*)d_in[10];
  const float* bias_out = (const float*)d_in[11];
  float* outp = (float*)d_out;
  (void)in_sizes; (void)n_in; (void)out_size; (void)d_ws; (void)ws_size;

  const size_t lds_bytes = (size_t)LDS_FLOATS * sizeof(float);  // ~252 KB
  hipFuncSetAttribute((const void*)ggnn_fused,
                      hipFuncAttributeMaxDynamicSharedMemorySize,
                      (int)lds_bytes);
  ggnn_fused<<<1024, 256, lds_bytes, stream>>>(
      A, hidden, w_ih, w_hh, b_ih, b_hh, b_iah, b_oah,
      W_in, bias_in, W_out, bias_out, outp);
}